// rAdjConv_10075993276647
// MI455X (gfx1250) — compile-verified
//
#include <hip/hip_runtime.h>
#include <hip/hip_bf16.h>

#define TILE    128      // edges staged per async tile (src/dst/div: 512 B each)
#define THREADS 256      // 8 waves; 16 groups of 16 lanes -> 16 edges in flight
#define DFEAT   64

// ---- CDNA5 inline-asm helpers -------------------------------------------

// Async DMA: global -> LDS, 16 bytes per active lane. Tracked by ASYNCcnt.
__device__ __forceinline__ void async_ld_b128(unsigned lds_byte_off, const void* gaddr) {
    asm volatile("global_load_async_to_lds_b128 %0, %1, off"
                 : : "v"(lds_byte_off), "v"(gaddr) : "memory");
}

template <int N>
__device__ __forceinline__ void wait_asynccnt() {
    asm volatile("s_wait_asynccnt %0" : : "i"(N) : "memory");
}

// Native non-returning f32 atomic add (L2-side RMW, no CAS loop, no return).
__device__ __forceinline__ void gatomic_add_f32(float* p, float v) {
    asm volatile("global_atomic_add_f32 %0, %1, off" : : "v"(p), "v"(v) : "memory");
}

// Generic pointer to a __shared__ object: low 32 bits are the LDS byte offset.
__device__ __forceinline__ unsigned lds_off(const void* p) {
    return (unsigned)(unsigned long long)p;
}

// ---- kernel --------------------------------------------------------------

__global__ __launch_bounds__(THREADS)
void rAdjConv_scatter_kernel(const float* __restrict__ x,
                             const int*   __restrict__ esrc,
                             const int*   __restrict__ edst,
                             const float* __restrict__ ediv,
                             float*       __restrict__ out,
                             int nEdges)
{
    __shared__ int   sSrc[2][TILE];
    __shared__ int   sDst[2][TILE];
    __shared__ float sDiv[2][TILE];

    const int tid  = threadIdx.x;
    const int lane = tid & 15;   // column chunk: 4 floats -> 16 lanes cover 64
    const int grp  = tid >> 4;   // 0..15 edge slot within a tile pass

    const int nFullTiles = nEdges / TILE;
    const int tstride    = gridDim.x;
    const bool issuer    = (tid < 32);   // wave 0 owns the async pipeline

    int t = blockIdx.x;

    // Prologue: stage first tile into buffer 0.
    if (t < nFullTiles && issuer) {
        const unsigned lb = (unsigned)tid * 16u;   // 32 lanes x 16B = 512B = TILE*4
        async_ld_b128(lds_off(&sSrc[0][0]) + lb, (const char*)(esrc + (size_t)t * TILE) + lb);
        async_ld_b128(lds_off(&sDst[0][0]) + lb, (const char*)(edst + (size_t)t * TILE) + lb);
        async_ld_b128(lds_off(&sDiv[0][0]) + lb, (const char*)(ediv + (size_t)t * TILE) + lb);
    }

    int buf = 0;
    for (; t < nFullTiles; t += tstride, buf ^= 1) {
        const int tn = t + tstride;
        if (issuer) {
            if (tn < nFullTiles) {
                // Issue next tile into the other buffer, then wait until only
                // those 3 new ops remain outstanding (ASYNCcnt completes
                // in order, so <=3 means the current tile's 3 ops are done).
                const unsigned lb = (unsigned)tid * 16u;
                const int nb = buf ^ 1;
                async_ld_b128(lds_off(&sSrc[nb][0]) + lb, (const char*)(esrc + (size_t)tn * TILE) + lb);
                async_ld_b128(lds_off(&sDst[nb][0]) + lb, (const char*)(edst + (size_t)tn * TILE) + lb);
                async_ld_b128(lds_off(&sDiv[nb][0]) + lb, (const char*)(ediv + (size_t)tn * TILE) + lb);
                wait_asynccnt<3>();
            } else {
                wait_asynccnt<0>();
            }
        }
        __syncthreads();   // publish staged tile to all 8 waves

        // 16 groups x 8 passes = 128 edges. Each group: 16 lanes x float4 = 1 row.
        #pragma unroll
        for (int k = 0; k < TILE / 16; ++k) {
            const int   e     = grp + k * 16;
            const int   s     = sSrc[buf][e];
            const int   d     = sDst[buf][e];
            const float scale = 1.0f / sDiv[buf][e];

            const float4 v = *(const float4*)(x + (size_t)s * DFEAT + (lane << 2));
            float* o = out + (size_t)d * DFEAT + (lane << 2);
            gatomic_add_f32(o + 0, v.x * scale);
            gatomic_add_f32(o + 1, v.y * scale);
            gatomic_add_f32(o + 2, v.z * scale);
            gatomic_add_f32(o + 3, v.w * scale);
        }
        __syncthreads();   // all reads of buf done before it is re-staged
    }

    // Remainder edges (nEdges % TILE) — direct path, block 0 only.
    const int base = nFullTiles * TILE;
    if (blockIdx.x == 0 && base < nEdges) {
        for (int e = base + grp; e < nEdges; e += 16) {
            const int   s     = esrc[e];
            const int   d     = edst[e];
            const float scale = 1.0f / ediv[e];
            const float4 v = *(const float4*)(x + (size_t)s * DFEAT + (lane << 2));
            float* o = out + (size_t)d * DFEAT + (lane << 2);
            gatomic_add_f32(o + 0, v.x * scale);
            gatomic_add_f32(o + 1, v.y * scale);
            gatomic_add_f32(o + 2, v.z * scale);
            gatomic_add_f32(o + 3, v.w * scale);
        }
    }
}

// ---- launch --------------------------------------------------------------

extern "C" void kernel_launch(void* const* d_in, const int* in_sizes, int n_in,
                              void* d_out, int out_size, void* d_ws, size_t ws_size,
                              hipStream_t stream) {
    const float* x    = (const float*)d_in[0];
    const int*   esrc = (const int*)  d_in[1];
    const int*   edst = (const int*)  d_in[2];
    const float* ediv = (const float*)d_in[3];
    float*       out  = (float*)d_out;
    const int nEdges  = in_sizes[1];

    // Output must start at zero every call (harness poisons once, never restores).
    hipMemsetAsync(d_out, 0, (size_t)out_size * sizeof(float), stream);

    int nTiles = nEdges / TILE;
    int blocks = nTiles > 2048 ? 2048 : (nTiles > 0 ? nTiles : 1);

    rAdjConv_scatter_kernel<<<blocks, THREADS, 0, stream>>>(x, esrc, edst, ediv, out, nEdges);
}